// SupervisedCL_43602507989896
// MI455X (gfx1250) — compile-verified
//
#include <hip/hip_runtime.h>
#include <hip/hip_bf16.h>

// ---------------------------------------------------------------------------
// Supervised contrastive loss, fused.  fp32 WMMA (v_wmma_f32_16x16x4_f32)
// keeps full precision through exp(sim/0.1) while using the CDNA5 matrix pipe.
// sims (128 MB) is never materialized: exp-row-sums and the intra-class block
// sums are folded into the GEMM epilogue.  All 16 MB of normalized data stays
// L2-resident (192 MB L2), so the GEMM is matrix-pipe bound, not HBM bound.
// ---------------------------------------------------------------------------

typedef float v2f __attribute__((ext_vector_type(2)));
typedef float v8f __attribute__((ext_vector_type(8)));

#define BATCH   8
#define NCLS    100
#define KPROTO  20
#define DDIM    256
#define MROWS   (NCLS * KPROTO)     // 2000 rows per batch
#define INV_T   10.0f               // 1/TEMP
#define ALPHA_F 0.1f
#define ROWTILE 16
#define NTILES  (MROWS / ROWTILE)   // 125
#define LDS_STRIDE 260              // 256 + 4 dwords: conflict-free ds_load_b64
#define NWAVES  4

// ---------------------------------------------------------------------------
// Kernel 1: L2-normalize each [D]-row; store xn and self_sim = ||xn||^2 / T.
// One wave32 per row, 8 floats per lane.
// ---------------------------------------------------------------------------
__global__ void __launch_bounds__(256)
cl_normalize(const float* __restrict__ x, float* __restrict__ xn,
             float* __restrict__ selfsim) {
    const int wave = threadIdx.x >> 5;
    const int lane = threadIdx.x & 31;
    const int row  = blockIdx.x * 8 + wave;          // 0..15999
    const float* xr = x + (size_t)row * DDIM + lane * 8;

    float4 a0 = *(const float4*)(xr);
    float4 a1 = *(const float4*)(xr + 4);
    float ss = a0.x*a0.x + a0.y*a0.y + a0.z*a0.z + a0.w*a0.w
             + a1.x*a1.x + a1.y*a1.y + a1.z*a1.z + a1.w*a1.w;
#pragma unroll
    for (int off = 16; off >= 1; off >>= 1) ss += __shfl_xor(ss, off, 32);

    const float inv = 1.0f / sqrtf(ss);
    float4 b0, b1;
    b0.x = a0.x*inv; b0.y = a0.y*inv; b0.z = a0.z*inv; b0.w = a0.w*inv;
    b1.x = a1.x*inv; b1.y = a1.y*inv; b1.z = a1.z*inv; b1.w = a1.w*inv;

    float* xw = xn + (size_t)row * DDIM + lane * 8;
    *(float4*)(xw)     = b0;
    *(float4*)(xw + 4) = b1;

    // self_sim recomputed from stored xn values (matches reference math)
    float s2 = b0.x*b0.x + b0.y*b0.y + b0.z*b0.z + b0.w*b0.w
             + b1.x*b1.x + b1.y*b1.y + b1.z*b1.z + b1.w*b1.w;
#pragma unroll
    for (int off = 16; off >= 1; off >>= 1) s2 += __shfl_xor(s2, off, 32);
    if (lane == 0) selfsim[row] = s2 * INV_T;
}

// ---------------------------------------------------------------------------
// Kernel 2: fused GEMM (Xn Xn^T / T) + exp-row-sum + intra-class sum.
// grid = (125 row tiles, 8 batches), block = 128 threads = 4 waves.
// A-tile (16 x 256 f32) staged in LDS (stride 260 dwords -> 64-bank perfect).
// Each wave strides over the 125 column tiles, 64 wmma_f32_16x16x4 per tile.
// ---------------------------------------------------------------------------
__global__ void __launch_bounds__(128)
cl_gemm_fused(const float* __restrict__ xn, const float* __restrict__ selfsim,
              float* __restrict__ partials) {
    __shared__ float ldsA[ROWTILE * LDS_STRIDE];
    __shared__ float redExp[NWAVES][ROWTILE];
    __shared__ float redIntra[NWAVES][ROWTILE];

    const int tid     = threadIdx.x;
    const int wave    = tid >> 5;
    const int lane    = tid & 31;
    const int b       = blockIdx.y;
    const int rowBase = blockIdx.x * ROWTILE;        // row tile in [0,2000)
    const float* xnb  = xn + (size_t)b * MROWS * DDIM;

    // Cooperative load of the 16x256 A-tile into LDS (float4 granularity).
#pragma unroll
    for (int i = tid; i < ROWTILE * (DDIM / 4); i += 128) {
        const int r  = i >> 6;           // /64 float4s per row
        const int c4 = i & 63;
        float4 v = *(const float4*)(xnb + (size_t)(rowBase + r) * DDIM + c4 * 4);
        *(float4*)(&ldsA[r * LDS_STRIDE + c4 * 4]) = v;
    }
    __syncthreads();

    // Per-lane fragment geometry (A and B share the same per-lane pattern).
    const int nLo   = lane & 15;          // column-in-tile (B) / row (A)
    const int kOff  = (lane >> 4) << 1;   // K sub-offset {0,2}
    const int mHalf = (lane >> 4) << 3;   // C rows handled: mHalf + v

    // Class block start column for each of this lane's 8 rows (tile-invariant).
    int c0v[8];
#pragma unroll
    for (int v = 0; v < 8; ++v)
        c0v[v] = ((rowBase + mHalf + v) / KPROTO) * KPROTO;

    const float* aLds = &ldsA[nLo * LDS_STRIDE + kOff];

    float expAcc[8], intraAcc[8];
#pragma unroll
    for (int v = 0; v < 8; ++v) { expAcc[v] = 0.0f; intraAcc[v] = 0.0f; }

    for (int jt = wave; jt < NTILES; jt += NWAVES) {
        const int colb = jt * ROWTILE;
        const float* bPtr = xnb + (size_t)(colb + nLo) * DDIM + kOff;

        v8f c = {0.f, 0.f, 0.f, 0.f, 0.f, 0.f, 0.f, 0.f};
#pragma unroll 8
        for (int d = 0; d < DDIM; d += 4) {
            v2f af = *(const v2f*)(aLds + d);     // ds_load_b64
            v2f bf = *(const v2f*)(bPtr + d);     // global_load_b64 (L2 hit)
            c = __builtin_amdgcn_wmma_f32_16x16x4_f32(
                    false, af, false, bf, (short)0, c, false, false);
        }

        // Epilogue: scale by 1/T, exp-accumulate, intra-class raw accumulate.
        const int col = colb + nLo;
#pragma unroll
        for (int v = 0; v < 8; ++v) {
            const float s = c[v] * INV_T;
            expAcc[v] += __expf(s);
            if (col >= c0v[v] && col < c0v[v] + KPROTO) intraAcc[v] += s;
        }
    }

    // Reduce across the 16 lanes of each half-wave (C layout: lanes 0-15 hold
    // rows mHalf=0..7, lanes 16-31 hold rows 8..15, one column per lane).
#pragma unroll
    for (int v = 0; v < 8; ++v) {
        float e = expAcc[v], it = intraAcc[v];
#pragma unroll
        for (int off = 8; off >= 1; off >>= 1) {
            e  += __shfl_xor(e,  off, 16);
            it += __shfl_xor(it, off, 16);
        }
        expAcc[v] = e; intraAcc[v] = it;
    }
    if ((lane & 15) == 0) {
#pragma unroll
        for (int v = 0; v < 8; ++v) {
            redExp[wave][mHalf + v]   = expAcc[v];
            redIntra[wave][mHalf + v] = intraAcc[v];
        }
    }
    __syncthreads();

    // First 16 threads: combine waves, form per-row loss terms, reduce.
    if (tid < 16) {
        float E = 0.0f, I = 0.0f;
#pragma unroll
        for (int w = 0; w < NWAVES; ++w) { E += redExp[w][tid]; I += redIntra[w][tid]; }
        const int gm = rowBase + tid;
        const float self  = selfsim[b * MROWS + gm];
        const float denom = E - __expf(self);
        float term = __logf(denom) - (I - self) * (1.0f / (KPROTO - 1));
#pragma unroll
        for (int off = 8; off >= 1; off >>= 1) term += __shfl_xor(term, off, 32);
        if (tid == 0) partials[b * NTILES + blockIdx.x] = term;
    }
}

// ---------------------------------------------------------------------------
// Kernel 3: deterministic tree reduction of the 1000 partials -> scalar loss.
// ---------------------------------------------------------------------------
__global__ void __launch_bounds__(256)
cl_finalize(const float* __restrict__ partials, float* __restrict__ out) {
    __shared__ float red[256];
    float s = 0.0f;
    for (int i = threadIdx.x; i < BATCH * NTILES; i += 256) s += partials[i];
    red[threadIdx.x] = s;
    __syncthreads();
#pragma unroll
    for (int off = 128; off >= 1; off >>= 1) {
        if (threadIdx.x < off) red[threadIdx.x] += red[threadIdx.x + off];
        __syncthreads();
    }
    if (threadIdx.x == 0)
        out[0] = red[0] * (ALPHA_F / (float)(1 * NCLS * KPROTO * DDIM));
}

// ---------------------------------------------------------------------------
extern "C" void kernel_launch(void* const* d_in, const int* in_sizes, int n_in,
                              void* d_out, int out_size, void* d_ws, size_t ws_size,
                              hipStream_t stream) {
    const float* x = (const float*)d_in[0];
    float* out = (float*)d_out;

    float* xn       = (float*)d_ws;                       // 8*2000*256 = 4,096,000 f
    float* selfsim  = xn + (size_t)BATCH * MROWS * DDIM;  // 16,000 f
    float* partials = selfsim + BATCH * MROWS;            // 1,000 f

    cl_normalize<<<dim3(BATCH * MROWS / 8), 256, 0, stream>>>(x, xn, selfsim);
    cl_gemm_fused<<<dim3(NTILES, BATCH), 128, 0, stream>>>(xn, selfsim, partials);
    cl_finalize<<<1, 256, 0, stream>>>(partials, out);
}